// Model_59906203845141
// MI455X (gfx1250) — compile-verified
//
#include <hip/hip_runtime.h>
#include <math.h>

// ---------------- model dims ----------------
#define L   256
#define D   300
#define Dp  304      // D padded to 16
#define S   16
#define KK  16
#define R   6
#define E   2048
#define C   7
#define Cp  16
#define D3  912      // 3 * Dp (segment-padded 3*D)
#define NN  48       // 3*S nodes per utterance
#define NEGF (-1000000000.0f)

typedef __attribute__((ext_vector_type(2))) float v2f;
typedef __attribute__((ext_vector_type(8))) float v8f;

// ---------------- workspace layout (floats) ----------------
constexpr size_t O_U      = 0;
constexpr size_t O_WREL   = O_U      + (size_t)L*Dp;        // R*Dp*Dp (B^T)
constexpr size_t O_WSELF  = O_WREL   + (size_t)R*Dp*Dp;     // Dp*Dp
constexpr size_t O_WQ     = O_WSELF  + (size_t)Dp*Dp;
constexpr size_t O_WK     = O_WQ     + (size_t)Dp*Dp;
constexpr size_t O_WV     = O_WK     + (size_t)Dp*Dp;
constexpr size_t O_WO     = O_WV     + (size_t)Dp*Dp;
constexpr size_t O_WOUT   = O_WO     + (size_t)Dp*Dp;       // Cp*Dp
constexpr size_t O_WFUSE  = O_WOUT   + (size_t)Cp*Dp;       // Dp*D3
constexpr size_t O_P      = O_WFUSE  + (size_t)Dp*D3;       // R*L*Dp
constexpr size_t O_HSELF  = O_P      + (size_t)R*L*Dp;
constexpr size_t O_HID    = O_HSELF  + (size_t)L*Dp;
constexpr size_t O_QM     = O_HID    + (size_t)L*Dp;
constexpr size_t O_KM     = O_QM     + (size_t)L*Dp;
constexpr size_t O_VM     = O_KM     + (size_t)L*Dp;
constexpr size_t O_AV     = O_VM     + (size_t)L*Dp;
constexpr size_t O_RELATT = O_AV     + (size_t)L*Dp;
constexpr size_t O_UNORM  = O_RELATT + (size_t)L*Dp;        // L
constexpr size_t O_NODES  = O_UNORM  + (size_t)L;           // L*NN*Dp
constexpr size_t O_NMASK  = O_NODES  + (size_t)L*NN*Dp;     // L*NN
constexpr size_t O_FEAT   = O_NMASK  + (size_t)L*NN;        // L*D3
constexpr size_t O_FUSE   = O_FEAT   + (size_t)L*D3;        // L*Dp
constexpr size_t O_H1     = O_FUSE   + (size_t)L*Dp;
constexpr size_t O_LOGITS = O_H1     + (size_t)L*Dp;        // L*Cp
constexpr size_t WS_TOTAL = O_LOGITS + (size_t)L*Cp;

// ---------------- helpers ----------------
__device__ __forceinline__ float wred(float v) {
#pragma unroll
  for (int o = 16; o > 0; o >>= 1) v += __shfl_down(v, o, 32);
  return v;
}

// ---------------- zero / pack kernels ----------------
__global__ void zero_k(float* p, size_t n) {
  size_t i = (size_t)blockIdx.x * blockDim.x + threadIdx.x;
  size_t st = (size_t)gridDim.x * blockDim.x;
  for (; i < n; i += st) p[i] = 0.0f;
}

__global__ void pack_u(const float* __restrict__ utt, float* __restrict__ up) {
  int i = blockIdx.x * blockDim.x + threadIdx.x;
  if (i >= L * D) return;
  int l = i / D, d = i % D;
  up[(size_t)l * Dp + d] = utt[i];
}

// W_rel[r][i][o] = comp[r,0]*Wb[0,i,o] + comp[r,1]*Wb[1,i,o]; store as B^T (o-major)
__global__ void pack_wrel(const float* __restrict__ Wb, const float* __restrict__ comp,
                          float* __restrict__ bt) {
  int i = blockIdx.x * blockDim.x + threadIdx.x;
  if (i >= R * D * D) return;
  int r = i / (D * D);
  int rem = i % (D * D);
  int in = rem / D, o = rem % D;
  float v = comp[r * 2 + 0] * Wb[(size_t)in * D + o] +
            comp[r * 2 + 1] * Wb[(size_t)D * D + (size_t)in * D + o];
  bt[(size_t)r * Dp * Dp + (size_t)o * Dp + in] = v;
}

// src is (K0 x N0) row-major; bt is (N x Kp) row-major (transposed, zero-padded)
__global__ void pack_bt(const float* __restrict__ src, float* __restrict__ bt,
                        int K0, int N0, int Kp) {
  int i = blockIdx.x * blockDim.x + threadIdx.x;
  if (i >= K0 * N0) return;
  int k = i / N0, n = i % N0;
  bt[(size_t)n * Kp + k] = src[i];
}

// W_fuse (900 x 300): rows remapped segment-wise into Kp=912
__global__ void pack_wfuse(const float* __restrict__ src, float* __restrict__ bt) {
  int i = blockIdx.x * blockDim.x + threadIdx.x;
  if (i >= 3 * D * D) return;
  int kin = i / D, o = i % D;
  int seg = kin / D, w = kin % D;
  bt[(size_t)o * D3 + (size_t)seg * Dp + w] = src[i];
}

// ---------------- WMMA f32 GEMM: C[M x N] = A[M x Kp] * B[Kp x N] (B stored B^T: N x Kp) ----------------
__global__ __launch_bounds__(32) void gemm16(
    const float* __restrict__ A, int lda, long sA,
    const float* __restrict__ BT, int ldb, long sB,
    float* __restrict__ Cm, int ldc, long sC, int Kp) {
  const int mBase = blockIdx.x * 16;
  const int nBase = blockIdx.y * 16;
  const float* Ab = A + (size_t)blockIdx.z * sA;
  const float* Bb = BT + (size_t)blockIdx.z * sB;
  float* Cb = Cm + (size_t)blockIdx.z * sC;

  const int lane = threadIdx.x;
  const int r = lane & 15;        // M row (for A) / N col (for B)
  const int half = lane >> 4;     // selects K pair {0,1} vs {2,3}

  const float* ap = Ab + (size_t)(mBase + r) * lda + 2 * half;
  const float* bp = Bb + (size_t)(nBase + r) * ldb + 2 * half;

  v8f acc = {0.f, 0.f, 0.f, 0.f, 0.f, 0.f, 0.f, 0.f};
  for (int k = 0; k < Kp; k += 4) {
    v2f a; a.x = ap[k]; a.y = ap[k + 1];
    v2f b; b.x = bp[k]; b.y = bp[k + 1];
    acc = __builtin_amdgcn_wmma_f32_16x16x4_f32(
        false, a, false, b, (short)0, acc, false, false);
  }
  float* cp = Cb + (size_t)(mBase + half * 8) * ldc + nBase + r;
#pragma unroll
  for (int g = 0; g < 8; ++g) cp[(size_t)g * ldc] = acc[g];
}

// ---------------- RGCN ----------------
__global__ void rgcn_edges(const float* __restrict__ P,
                           const int* __restrict__ src, const int* __restrict__ dst,
                           const int* __restrict__ et, float* __restrict__ hid) {
  int e = blockIdx.x;
  int t = threadIdx.x;
  int s = src[e], d0 = dst[e], rr = et[e];
  const float* row = P + ((size_t)rr * L + s) * Dp;
  float* out = hid + (size_t)d0 * Dp;
  for (int d = t; d < D; d += 64) atomicAdd(&out[d], row[d]);
}

__global__ void rgcn_finish(float* __restrict__ hid, const float* __restrict__ hs,
                            const float* __restrict__ b) {
  int i = blockIdx.x * blockDim.x + threadIdx.x;
  if (i >= L * D) return;
  int l = i / D, d = i % D;
  hid[(size_t)l * Dp + d] += hs[(size_t)l * Dp + d] + b[d];
}

// ---------------- windowed attention (one wave per (l, head)) ----------------
__global__ __launch_bounds__(32) void attn_win(const float* __restrict__ Q,
                                               const float* __restrict__ Km,
                                               const float* __restrict__ Vm,
                                               float* __restrict__ AV) {
  int l = blockIdx.x, h = blockIdx.y, lane = threadIdx.x;
  int base = h * (D / 2);
  int j[3]; float sc[3];
#pragma unroll
  for (int w = 0; w < 3; ++w) {
    int jj = l - 1 + w;
    jj = jj < 0 ? 0 : (jj > L - 1 ? L - 1 : jj);
    j[w] = jj;
    float p = 0.f;
    for (int d = lane; d < D / 2; d += 32)
      p += Q[(size_t)l * Dp + base + d] * Km[(size_t)jj * Dp + base + d];
    p = wred(p);
    p = __shfl(p, 0, 32);
    sc[w] = p * 0.0816496581f;   // 1/sqrt(150)
  }
  float mx = fmaxf(sc[0], fmaxf(sc[1], sc[2]));
  float e0 = expf(sc[0] - mx), e1 = expf(sc[1] - mx), e2 = expf(sc[2] - mx);
  float inv = 1.0f / (e0 + e1 + e2);
  for (int d = lane; d < D / 2; d += 32)
    AV[(size_t)l * Dp + base + d] =
        (e0 * Vm[(size_t)j[0] * Dp + base + d] +
         e1 * Vm[(size_t)j[1] * Dp + base + d] +
         e2 * Vm[(size_t)j[2] * Dp + base + d]) * inv;
}

__global__ __launch_bounds__(32) void unorm_k(const float* __restrict__ rel,
                                              float* __restrict__ un) {
  int l = blockIdx.x, lane = threadIdx.x;
  float p = 0.f;
  for (int d = lane; d < D; d += 32) {
    float v = rel[(size_t)l * Dp + d];
    p += v * v;
  }
  p = wred(p);
  if (lane == 0) un[l] = sqrtf(p);
}

// ---------------- concept module: one wave per (g, l, s) ----------------
__global__ __launch_bounds__(32) void concept_k(
    const float* __restrict__ rel, const float* __restrict__ un,
    const float* __restrict__ table,
    const int* __restrict__ srcIds, const int* __restrict__ dstIds,
    const float* __restrict__ wgt, const float* __restrict__ sentic,
    const float* __restrict__ rvecs,
    float* __restrict__ nodes, float* __restrict__ nmask) {
  const int s = blockIdx.x, l = blockIdx.y, g = blockIdx.z;
  const int lane = threadIdx.x;

  __shared__ float sdst[KK][Dp];
  __shared__ float ssrc[Dp], srw[Dp];
  __shared__ float sd1[KK], sd2[KK], sn[KK];
  __shared__ int smask[KK];

  const size_t sidx = ((size_t)g * L + l) * S + s;
  const int srcId = srcIds[sidx];
  const int srcRow = srcId < 0 ? 0 : srcId;
  const float* u = rel + (size_t)l * Dp;
  const float unv = un[l];

  for (int d = lane; d < D; d += 32) {
    float sv = table[(size_t)srcRow * D + d];
    ssrc[d] = sv;
    srw[d] = sv * rvecs[g * D + d];
  }
  __syncthreads();

  for (int k = 0; k < KK; ++k) {
    const size_t didx = sidx * KK + k;
    const int dId = dstIds[didx];
    const int dRow = dId < 0 ? 0 : dId;
    if (k + 1 < KK) {  // prefetch next gather row (global_prefetch)
      int nId = dstIds[didx + 1];
      const float* np = table + (size_t)(nId < 0 ? 0 : nId) * D;
      __builtin_prefetch(np + lane * 8, 0, 1);
    }
    float p1 = 0.f, p2 = 0.f, pn = 0.f;
    for (int d = lane; d < D; d += 32) {
      float dv = table[(size_t)dRow * D + d];
      sdst[k][d] = dv;
      p1 += u[d] * dv;
      p2 += srw[d] * dv;
      pn += dv * dv;
    }
    p1 = wred(p1); p2 = wred(p2); pn = wred(pn);
    if (lane == 0) { sd1[k] = p1; sd2[k] = p2; sn[k] = pn; smask[k] = (dId >= 0); }
  }
  __syncthreads();

  // per-lane redundant scalar path (all identical values)
  float omv[KK], a1[KK], s2[KK], cvals[KK];
  float mx1 = NEGF;
#pragma unroll
  for (int k = 0; k < KK; ++k) {
    const size_t didx = sidx * KK + k;
    float o;
    if (smask[k]) {
      float cosv = fabsf(sd1[k]) / (unv * sqrtf(sn[k]) + 1e-8f);
      o = 0.5f * wgt[didx] * cosv + 0.5f * fabsf(sentic[didx]);
    } else o = NEGF;
    omv[k] = o;
    mx1 = fmaxf(mx1, o);
  }
  float sum1 = 0.f;
#pragma unroll
  for (int k = 0; k < KK; ++k) { float e = expf(omv[k] - mx1); a1[k] = e; sum1 += e; }
  float inv1 = 1.0f / sum1;
#pragma unroll
  for (int k = 0; k < KK; ++k) a1[k] = a1[k] * inv1 * (smask[k] ? 1.f : 0.f);

  float mx2 = NEGF;
#pragma unroll
  for (int k = 0; k < KK; ++k) {
    float v = smask[k] ? a1[k] * sd2[k] : NEGF;
    s2[k] = v;
    mx2 = fmaxf(mx2, v);
  }
  float sum2 = 0.f;
#pragma unroll
  for (int k = 0; k < KK; ++k) { float e = expf(s2[k] - mx2); s2[k] = e; sum2 += e; }
  float inv2 = 1.0f / sum2;
#pragma unroll
  for (int k = 0; k < KK; ++k)
    cvals[k] = a1[k] * (s2[k] * inv2 * (smask[k] ? 1.f : 0.f));

  float* out = nodes + ((size_t)l * NN + g * S + s) * Dp;
  for (int d = lane; d < D; d += 32) {
    float acc = 0.f;
#pragma unroll
    for (int k = 0; k < KK; ++k) acc += cvals[k] * sdst[k][d];
    out[d] = ssrc[d] + rvecs[g * D + d] * acc;
  }
  if (lane == 0) nmask[(size_t)l * NN + g * S + s] = (srcId >= 0) ? 1.f : 0.f;
}

// ---------------- symbolic attention over 48 nodes ----------------
__global__ __launch_bounds__(64) void sym_k(const float* __restrict__ nodes,
                                            const float* __restrict__ nmask,
                                            const float* __restrict__ rel,
                                            float* __restrict__ feat) {
  int l = blockIdx.x;
  int tid = threadIdx.x, wave = tid >> 5, lane = tid & 31;
  __shared__ float sc[NN];
  __shared__ float att[NN];
  const float* u = rel + (size_t)l * Dp;
  for (int n = wave; n < NN; n += 2) {
    const float* nd = nodes + ((size_t)l * NN + n) * Dp;
    float p = 0.f;
    for (int d = lane; d < D; d += 32) p += nd[d] * u[d];
    p = wred(p);
    if (lane == 0) sc[n] = p;
  }
  __syncthreads();
  if (tid == 0) {
    float mx = NEGF;
    for (int n = 0; n < NN; ++n) {
      float v = (nmask[(size_t)l * NN + n] > 0.f) ? sc[n] : NEGF;
      sc[n] = v;
      mx = fmaxf(mx, v);
    }
    float sum = 0.f;
    for (int n = 0; n < NN; ++n) {
      float e = expf(sc[n] - mx);
      att[n] = e * nmask[(size_t)l * NN + n];
      sum += e;
    }
    float inv = 1.0f / sum;
    for (int n = 0; n < NN; ++n) att[n] *= inv;
  }
  __syncthreads();
  for (int d = tid; d < D; d += 64) {
    float acc = 0.f;
    for (int n = 0; n < NN; ++n)
      acc += att[n] * nodes[((size_t)l * NN + n) * Dp + d];
    feat[(size_t)l * D3 + 2 * Dp + d] = acc;
  }
}

__global__ void feat_pack(const float* __restrict__ hid, const float* __restrict__ rel,
                          float* __restrict__ feat) {
  int i = blockIdx.x * blockDim.x + threadIdx.x;
  if (i >= L * D) return;
  int l = i / D, d = i % D;
  feat[(size_t)l * D3 + d] = hid[(size_t)l * Dp + d];
  feat[(size_t)l * D3 + Dp + d] = rel[(size_t)l * Dp + d];
}

__global__ void epi1(const float* __restrict__ fuse, const float* __restrict__ bf,
                     float* __restrict__ h1) {
  int i = blockIdx.x * blockDim.x + threadIdx.x;
  if (i >= L * D) return;
  int l = i / D, d = i % D;
  h1[(size_t)l * Dp + d] = fmaxf(fuse[(size_t)l * Dp + d] + bf[d], 0.f);
}

__global__ void final_k(const float* __restrict__ logits, const float* __restrict__ bo,
                        float* __restrict__ out) {
  int l = blockIdx.x * blockDim.x + threadIdx.x;
  if (l >= L) return;
  float v[C];
  float mx = NEGF;
#pragma unroll
  for (int c = 0; c < C; ++c) {
    v[c] = logits[(size_t)l * Cp + c] + bo[c];
    mx = fmaxf(mx, v[c]);
  }
  float sum = 0.f;
#pragma unroll
  for (int c = 0; c < C; ++c) sum += expf(v[c] - mx);
  float lse = mx + logf(sum);
#pragma unroll
  for (int c = 0; c < C; ++c) out[(size_t)l * C + c] = v[c] - lse;
}

// ---------------- host launch ----------------
extern "C" void kernel_launch(void* const* d_in, const int* in_sizes, int n_in,
                              void* d_out, int out_size, void* d_ws, size_t ws_size,
                              hipStream_t stream) {
  const float* utt     = (const float*)d_in[0];
  const int*   str_src = (const int*)  d_in[1];
  const int*   str_dst = (const int*)  d_in[2];
  const int*   str_et  = (const int*)  d_in[3];
  const int*   cSrc    = (const int*)  d_in[4];
  const int*   cDst    = (const int*)  d_in[5];
  const float* cWgt    = (const float*)d_in[6];
  const float* cSen    = (const float*)d_in[7];
  const float* table   = (const float*)d_in[8];
  const float* Wbasis  = (const float*)d_in[9];
  const float* comp    = (const float*)d_in[10];
  const float* Wself   = (const float*)d_in[11];
  const float* b_rgcn  = (const float*)d_in[12];
  const float* Wq      = (const float*)d_in[13];
  const float* Wk      = (const float*)d_in[14];
  const float* Wv      = (const float*)d_in[15];
  const float* Wo      = (const float*)d_in[16];
  const float* rvecs   = (const float*)d_in[17];
  const float* Wfuse   = (const float*)d_in[18];
  const float* bfuse   = (const float*)d_in[19];
  const float* Wout    = (const float*)d_in[20];
  const float* bout    = (const float*)d_in[21];
  float* out = (float*)d_out;
  float* ws = (float*)d_ws;

  const int TPB = 256;
  auto nb = [](long n, int t) { return (int)((n + t - 1) / t); };

  // 0) zero workspace (every call: deterministic, covers poison)
  zero_k<<<4096, TPB, 0, stream>>>(ws, WS_TOTAL);

  // 1) packing
  pack_u   <<<nb((long)L * D, TPB), TPB, 0, stream>>>(utt, ws + O_U);
  pack_wrel<<<nb((long)R * D * D, TPB), TPB, 0, stream>>>(Wbasis, comp, ws + O_WREL);
  pack_bt  <<<nb((long)D * D, TPB), TPB, 0, stream>>>(Wself, ws + O_WSELF, D, D, Dp);
  pack_bt  <<<nb((long)D * D, TPB), TPB, 0, stream>>>(Wq, ws + O_WQ, D, D, Dp);
  pack_bt  <<<nb((long)D * D, TPB), TPB, 0, stream>>>(Wk, ws + O_WK, D, D, Dp);
  pack_bt  <<<nb((long)D * D, TPB), TPB, 0, stream>>>(Wv, ws + O_WV, D, D, Dp);
  pack_bt  <<<nb((long)D * D, TPB), TPB, 0, stream>>>(Wo, ws + O_WO, D, D, Dp);
  pack_bt  <<<nb((long)D * C, TPB), TPB, 0, stream>>>(Wout, ws + O_WOUT, D, C, Dp);
  pack_wfuse<<<nb((long)3 * D * D, TPB), TPB, 0, stream>>>(Wfuse, ws + O_WFUSE);

  // 2) GEMMs on the WMMA unit
  dim3 gLN(L / 16, Dp / 16, 1);
  gemm16<<<gLN, 32, 0, stream>>>(ws + O_U, Dp, 0, ws + O_WSELF, Dp, 0,
                                 ws + O_HSELF, Dp, 0, Dp);
  gemm16<<<dim3(L / 16, Dp / 16, R), 32, 0, stream>>>(
      ws + O_U, Dp, 0, ws + O_WREL, Dp, (long)Dp * Dp,
      ws + O_P, Dp, (long)L * Dp, Dp);
  gemm16<<<gLN, 32, 0, stream>>>(ws + O_U, Dp, 0, ws + O_WQ, Dp, 0, ws + O_QM, Dp, 0, Dp);
  gemm16<<<gLN, 32, 0, stream>>>(ws + O_U, Dp, 0, ws + O_WK, Dp, 0, ws + O_KM, Dp, 0, Dp);
  gemm16<<<gLN, 32, 0, stream>>>(ws + O_U, Dp, 0, ws + O_WV, Dp, 0, ws + O_VM, Dp, 0, Dp);

  // 3) RGCN aggregation
  rgcn_edges<<<E, 64, 0, stream>>>(ws + O_P, str_src, str_dst, str_et, ws + O_HID);
  rgcn_finish<<<nb((long)L * D, TPB), TPB, 0, stream>>>(ws + O_HID, ws + O_HSELF, b_rgcn);

  // 4) windowed attention -> AV -> @Wo -> relatt
  attn_win<<<dim3(L, 2, 1), 32, 0, stream>>>(ws + O_QM, ws + O_KM, ws + O_VM, ws + O_AV);
  gemm16<<<gLN, 32, 0, stream>>>(ws + O_AV, Dp, 0, ws + O_WO, Dp, 0,
                                 ws + O_RELATT, Dp, 0, Dp);
  unorm_k<<<L, 32, 0, stream>>>(ws + O_RELATT, ws + O_UNORM);

  // 5) concept module
  concept_k<<<dim3(S, L, 3), 32, 0, stream>>>(
      ws + O_RELATT, ws + O_UNORM, table, cSrc, cDst, cWgt, cSen, rvecs,
      ws + O_NODES, ws + O_NMASK);

  // 6) symbolic attention + feat assembly
  sym_k<<<L, 64, 0, stream>>>(ws + O_NODES, ws + O_NMASK, ws + O_RELATT, ws + O_FEAT);
  feat_pack<<<nb((long)L * D, TPB), TPB, 0, stream>>>(ws + O_HID, ws + O_RELATT, ws + O_FEAT);

  // 7) fuse GEMM (K=912), ReLU, out GEMM (N=16), log-softmax
  gemm16<<<dim3(L / 16, Dp / 16, 1), 32, 0, stream>>>(
      ws + O_FEAT, D3, 0, ws + O_WFUSE, D3, 0, ws + O_FUSE, Dp, 0, D3);
  epi1<<<nb((long)L * D, TPB), TPB, 0, stream>>>(ws + O_FUSE, bfuse, ws + O_H1);
  gemm16<<<dim3(L / 16, 1, 1), 32, 0, stream>>>(
      ws + O_H1, Dp, 0, ws + O_WOUT, Dp, 0, ws + O_LOGITS, Cp, 0, Dp);
  final_k<<<1, L, 0, stream>>>(ws + O_LOGITS, bout, out);
}